// TensorEmbedding_19808389169520
// MI455X (gfx1250) — compile-verified
//
#include <hip/hip_runtime.h>
#include <math.h>

// ---------------------------------------------------------------------------
// TensorNet-style embedding layer for MI455X (gfx1250), wave32 + f32 WMMA.
// All GEMMs use V_WMMA_F32_16X16X4_F32 (fp32 precision matches reference).
// ---------------------------------------------------------------------------

typedef __attribute__((ext_vector_type(2))) float v2f;
typedef __attribute__((ext_vector_type(8))) float v8f;

#define UDIM 32
#define RDIM 64
#define CUTOFF_F 5.0f
#define PI_F 3.14159265358979323846f

__device__ __forceinline__ v8f wmma4(v2f a, v2f b, v8f c) {
  // D = A(16x4 f32) * B(4x16 f32) + C(16x16 f32)
  return __builtin_amdgcn_wmma_f32_16x16x4_f32(
      /*neg_a=*/false, a, /*neg_b=*/false, b,
      /*c_mod=*/(short)0, c, /*reuse_a=*/false, /*reuse_b=*/false);
}

__device__ __forceinline__ v8f zero8() {
  v8f z = {0.f, 0.f, 0.f, 0.f, 0.f, 0.f, 0.f, 0.f};
  return z;
}

// ---------------------------------------------------------------------------
// Kernel 0: zero the per-node segment accumulators (segFI | segA | segS)
// ---------------------------------------------------------------------------
__global__ void te_zero_ws(float* ws, int n) {
  int i = blockIdx.x * blockDim.x + threadIdx.x;
  if (i < n) ws[i] = 0.f;
}

// ---------------------------------------------------------------------------
// Kernel 1: edge kernel. One wave = one 16-edge tile.
//   GEMMs (WMMA f32): edge_attr@{Wd1,Wd2,Wd3,Wemb3}, Zcat@Wemb2
//   Then fused cutoff / tensor components / atomic segment reduction.
// ---------------------------------------------------------------------------
__global__ __launch_bounds__(128) void te_edge_kernel(
    const float* __restrict__ edge_attr, const float* __restrict__ bond_dist,
    const float* __restrict__ bond_vec, const int* __restrict__ src,
    const int* __restrict__ dst, const int* __restrict__ node_type,
    const float* __restrict__ emb_table,
    const float* __restrict__ Wd1, const float* __restrict__ bd1,
    const float* __restrict__ Wd2, const float* __restrict__ bd2,
    const float* __restrict__ Wd3, const float* __restrict__ bd3,
    const float* __restrict__ Wemb2, const float* __restrict__ bemb2,
    const float* __restrict__ Wemb3, const float* __restrict__ bemb3,
    float* __restrict__ segFI, float* __restrict__ segA,
    float* __restrict__ segS, float* __restrict__ edge_feat,
    int E) {
  const int lane = threadIdx.x & 31;
  const int wave = threadIdx.x >> 5;
  const long tile = (long)blockIdx.x * (blockDim.x >> 5) + wave;
  const long tiles = ((long)E + 15) / 16;
  if (tile >= tiles) return;  // wave-uniform exit

  const long ebase = tile * 16;
  const int mrow = lane & 15;   // A row / B,D column within 16
  const int hh = lane >> 4;     // lane half
  const int kb = hh * 2;        // K sub-offset for A/B fragments

  long em = ebase + mrow;
  if (em >= E) em = E - 1;  // clamp reads; writes are predicated
  const int sm = src[em];
  const int dm = dst[em];
  const float* zs = emb_table + (long)node_type[sm] * UDIM;
  const float* zd = emb_table + (long)node_type[dm] * UDIM;
  const float* arow = edge_attr + em * (long)RDIM;

  v8f accW1[2] = {zero8(), zero8()};
  v8f accW2[2] = {zero8(), zero8()};
  v8f accW3[2] = {zero8(), zero8()};
  v8f accEF[2] = {zero8(), zero8()};
  v8f accZ[2]  = {zero8(), zero8()};

  // ---- GEMMs with edge_attr as A (shared A fragment across 4 weights) ----
  for (int k0 = 0; k0 < RDIM; k0 += 4) {
    v2f a;
    a.x = arow[k0 + kb];
    a.y = arow[k0 + kb + 1];
    #pragma unroll
    for (int nt = 0; nt < 2; ++nt) {
      const int n = nt * 16 + mrow;
      v2f b;
      b.x = Wd1[(k0 + kb) * UDIM + n];  b.y = Wd1[(k0 + kb + 1) * UDIM + n];
      accW1[nt] = wmma4(a, b, accW1[nt]);
      b.x = Wd2[(k0 + kb) * UDIM + n];  b.y = Wd2[(k0 + kb + 1) * UDIM + n];
      accW2[nt] = wmma4(a, b, accW2[nt]);
      b.x = Wd3[(k0 + kb) * UDIM + n];  b.y = Wd3[(k0 + kb + 1) * UDIM + n];
      accW3[nt] = wmma4(a, b, accW3[nt]);
      b.x = Wemb3[(k0 + kb) * UDIM + n]; b.y = Wemb3[(k0 + kb + 1) * UDIM + n];
      accEF[nt] = wmma4(a, b, accEF[nt]);
    }
  }

  // ---- Zij = concat(Z[src], Z[dst]) @ Wemb2 ----
  for (int k0 = 0; k0 < 2 * UDIM; k0 += 4) {
    const int k = k0 + kb;  // even
    v2f a;
    if (k < UDIM) { a.x = zs[k]; a.y = zs[k + 1]; }
    else          { a.x = zd[k - UDIM]; a.y = zd[k - UDIM + 1]; }
    #pragma unroll
    for (int nt = 0; nt < 2; ++nt) {
      const int n = nt * 16 + mrow;
      v2f b;
      b.x = Wemb2[k * UDIM + n];
      b.y = Wemb2[(k + 1) * UDIM + n];
      accZ[nt] = wmma4(a, b, accZ[nt]);
    }
  }

  // ---- epilogue: per D-row edge math + atomic segment reduction ----
  #pragma unroll
  for (int i = 0; i < 8; ++i) {
    const int r = i + 8 * hh;  // D-fragment row
    const long e = ebase + r;
    if (e >= E) continue;
    const float d = bond_dist[e];
    const float Cc = 0.5f * (cosf(PI_F * d / CUTOFF_F) + 1.f) *
                     (d <= CUTOFF_F ? 1.f : 0.f);
    float v0 = bond_vec[3 * e + 0];
    float v1 = bond_vec[3 * e + 1];
    float v2 = bond_vec[3 * e + 2];
    const float inv = rsqrtf(v0 * v0 + v1 * v1 + v2 * v2);
    v0 *= inv; v1 *= inv; v2 *= inv;
    const float third = 1.f / 3.f;
    const float sxx = v0 * v0 - third, syy = v1 * v1 - third;
    const float sxy = v0 * v1, sxz = v0 * v2, syz = v1 * v2;
    const int dn = dst[e];
    #pragma unroll
    for (int nt = 0; nt < 2; ++nt) {
      const int n = nt * 16 + mrow;
      const float w1 = (accW1[nt][i] + bd1[n]) * Cc;
      const float w2 = (accW2[nt][i] + bd2[n]) * Cc;
      const float w3 = (accW3[nt][i] + bd3[n]) * Cc;
      const float zij = accZ[nt][i] + bemb2[n];
      const float fI = zij * w1, fA = zij * w2, fS = zij * w3;
      atomicAdd(segFI + (long)dn * UDIM + n, fI);
      float* pA = segA + ((long)dn * UDIM + n) * 3;
      atomicAdd(pA + 0, fA * v0);
      atomicAdd(pA + 1, fA * v1);
      atomicAdd(pA + 2, fA * v2);
      float* pS = segS + ((long)dn * UDIM + n) * 5;
      atomicAdd(pS + 0, fS * sxx);
      atomicAdd(pS + 1, fS * sxy);
      atomicAdd(pS + 2, fS * sxz);
      atomicAdd(pS + 3, fS * syy);
      atomicAdd(pS + 4, fS * syz);
      edge_feat[e * UDIM + n] = accEF[nt][i] + bemb3[n];
    }
  }
}

// ---------------------------------------------------------------------------
// Kernel 2: node kernel. One wave = one 16-node tile. LDS-staged pipeline:
//   tensor-norm -> LayerNorm -> WMMA MLP (silu) -> 9 WMMA mixing GEMMs ->
//   reconstruct X[n,v,3,3].
// Per-wave LDS: nrm[16][32] | act1[16][64] | gact[16][96] | mix[16][288]
// ---------------------------------------------------------------------------
#define NODE_LDS_PER_WAVE (512 + 1024 + 1536 + 4608)  // 7680 floats

__global__ __launch_bounds__(64) void te_node_kernel(
    const float* __restrict__ segFI, const float* __restrict__ segA,
    const float* __restrict__ segS,
    const float* __restrict__ Wt0, const float* __restrict__ Wt1,
    const float* __restrict__ Wt2,
    const float* __restrict__ Ws1, const float* __restrict__ bs1,
    const float* __restrict__ Ws2, const float* __restrict__ bs2,
    const float* __restrict__ ln_g, const float* __restrict__ ln_b,
    float* __restrict__ X_out, int N) {
  __shared__ float lds[2 * NODE_LDS_PER_WAVE];  // 61,440 bytes

  const int lane = threadIdx.x & 31;
  const int wave = threadIdx.x >> 5;
  const long tile = (long)blockIdx.x * (blockDim.x >> 5) + wave;
  const long tiles = ((long)N + 15) / 16;
  if (tile >= tiles) return;

  const long nbase = tile * 16;
  const int mrow = lane & 15;
  const int hh = lane >> 4;
  const int kb = hh * 2;

  float* nrmT = lds + wave * NODE_LDS_PER_WAVE;  // [16][32]
  float* act1 = nrmT + 512;                      // [16][64]
  float* gact = act1 + 1024;                     // [16][96]
  float* mix  = gact + 1536;                     // [16][288]  (9 comps x 32)

  // ---- step 1: tensor norm per (node, u) ----
  for (int t = lane; t < 16 * UDIM; t += 32) {
    const int nd = t >> 5, u = t & 31;
    long node = nbase + nd;
    if (node >= N) node = N - 1;
    const long o = node * UDIM + u;
    const float fI = segFI[o];
    const float* a = segA + o * 3;
    const float* s = segS + o * 5;
    const float a0 = a[0], a1 = a[1], a2 = a[2];
    const float sxx = s[0], sxy = s[1], sxz = s[2], syy = s[3], syz = s[4];
    const float szz = -(sxx + syy);
    const float m00 = fI + sxx, m01 = sxy - a2, m02 = sxz + a1;
    const float m10 = sxy + a2, m11 = fI + syy, m12 = syz - a0;
    const float m20 = sxz - a1, m21 = syz + a0, m22 = fI + szz;
    nrmT[t] = m00 * m00 + m01 * m01 + m02 * m02 + m10 * m10 + m11 * m11 +
              m12 * m12 + m20 * m20 + m21 * m21 + m22 * m22;
  }
  __syncthreads();

  // ---- step 2: LayerNorm over U per node ----
  if (lane < 16) {
    float* row = nrmT + lane * UDIM;
    float mu = 0.f;
    for (int u = 0; u < UDIM; ++u) mu += row[u];
    mu *= (1.f / UDIM);
    float var = 0.f;
    for (int u = 0; u < UDIM; ++u) { const float dd = row[u] - mu; var += dd * dd; }
    var *= (1.f / UDIM);
    const float inv = rsqrtf(var + 1e-5f);
    for (int u = 0; u < UDIM; ++u)
      row[u] = (row[u] - mu) * inv * ln_g[u] + ln_b[u];
  }
  __syncthreads();

  // ---- step 3: MLP layer 1: [16,32] @ Ws1[32,64] + bias -> silu ----
  for (int nt = 0; nt < 4; ++nt) {
    v8f acc = zero8();
    for (int k0 = 0; k0 < UDIM; k0 += 4) {
      const int k = k0 + kb;
      v2f a; a.x = nrmT[mrow * UDIM + k]; a.y = nrmT[mrow * UDIM + k + 1];
      v2f b; b.x = Ws1[k * 64 + nt * 16 + mrow];
             b.y = Ws1[(k + 1) * 64 + nt * 16 + mrow];
      acc = wmma4(a, b, acc);
    }
    #pragma unroll
    for (int i = 0; i < 8; ++i) {
      const int r = i + 8 * hh;
      const int col = nt * 16 + mrow;
      const float x = acc[i] + bs1[col];
      act1[r * 64 + col] = x / (1.f + expf(-x));
    }
  }
  __syncthreads();

  // ---- step 4: MLP layer 2: [16,64] @ Ws2[64,96] + bias -> silu ----
  for (int nt = 0; nt < 6; ++nt) {
    v8f acc = zero8();
    for (int k0 = 0; k0 < 64; k0 += 4) {
      const int k = k0 + kb;
      v2f a; a.x = act1[mrow * 64 + k]; a.y = act1[mrow * 64 + k + 1];
      v2f b; b.x = Ws2[k * 96 + nt * 16 + mrow];
             b.y = Ws2[(k + 1) * 96 + nt * 16 + mrow];
      acc = wmma4(a, b, acc);
    }
    #pragma unroll
    for (int i = 0; i < 8; ++i) {
      const int r = i + 8 * hh;
      const int col = nt * 16 + mrow;
      const float x = acc[i] + bs2[col];
      gact[r * 96 + col] = x / (1.f + expf(-x));
    }
  }
  __syncthreads();

  // ---- step 5: channel-mixing einsums as 9 GEMMs [16,32] @ Wt[32,32] ----
  long nodeA = nbase + mrow;
  if (nodeA >= N) nodeA = N - 1;
  for (int c = 0; c < 9; ++c) {
    const float* Wt = (c == 0) ? Wt0 : ((c < 4) ? Wt1 : Wt2);
    for (int nt = 0; nt < 2; ++nt) {
      v8f acc = zero8();
      for (int k0 = 0; k0 < UDIM; k0 += 4) {
        const int k = k0 + kb;
        v2f a;
        if (c == 0) {
          a.x = segFI[nodeA * UDIM + k];
          a.y = segFI[nodeA * UDIM + k + 1];
        } else if (c < 4) {
          a.x = segA[(nodeA * UDIM + k) * 3 + (c - 1)];
          a.y = segA[(nodeA * UDIM + k + 1) * 3 + (c - 1)];
        } else {
          a.x = segS[(nodeA * UDIM + k) * 5 + (c - 4)];
          a.y = segS[(nodeA * UDIM + k + 1) * 5 + (c - 4)];
        }
        v2f b; b.x = Wt[k * UDIM + nt * 16 + mrow];
               b.y = Wt[(k + 1) * UDIM + nt * 16 + mrow];
        acc = wmma4(a, b, acc);
      }
      #pragma unroll
      for (int i = 0; i < 8; ++i) {
        const int r = i + 8 * hh;
        mix[r * 288 + c * 32 + nt * 16 + mrow] = acc[i];
      }
    }
  }
  __syncthreads();

  // ---- step 6: reconstruct X[n,v,3,3] ----
  for (int t = lane; t < 16 * UDIM; t += 32) {
    const int nd = t >> 5, v = t & 31;
    const long node = nbase + nd;
    if (node >= N) continue;
    const float* mr = mix + nd * 288;
    const float fI  = mr[0 * 32 + v];
    const float a0  = mr[1 * 32 + v], a1 = mr[2 * 32 + v], a2 = mr[3 * 32 + v];
    const float sxx = mr[4 * 32 + v], sxy = mr[5 * 32 + v], sxz = mr[6 * 32 + v];
    const float syy = mr[7 * 32 + v], syz = mr[8 * 32 + v];
    const float g0 = gact[nd * 96 + 3 * v + 0];
    const float g1 = gact[nd * 96 + 3 * v + 1];
    const float g2 = gact[nd * 96 + 3 * v + 2];
    float* o = X_out + (node * UDIM + v) * 9;
    o[0] = g0 * fI + g2 * sxx;
    o[1] = -g1 * a2 + g2 * sxy;
    o[2] = g1 * a1 + g2 * sxz;
    o[3] = g1 * a2 + g2 * sxy;
    o[4] = g0 * fI + g2 * syy;
    o[5] = -g1 * a0 + g2 * syz;
    o[6] = -g1 * a1 + g2 * sxz;
    o[7] = g1 * a0 + g2 * syz;
    o[8] = g0 * fI - g2 * (sxx + syy);
  }
}

// ---------------------------------------------------------------------------
// Launch wrapper
// ---------------------------------------------------------------------------
extern "C" void kernel_launch(void* const* d_in, const int* in_sizes, int n_in,
                              void* d_out, int out_size, void* d_ws,
                              size_t ws_size, hipStream_t stream) {
  const int* node_type  = (const int*)d_in[0];
  const float* edge_attr = (const float*)d_in[1];
  const float* bond_dist = (const float*)d_in[2];
  const float* bond_vec  = (const float*)d_in[3];
  const int* src = (const int*)d_in[4];
  const int* dst = (const int*)d_in[5];
  const float* emb_table = (const float*)d_in[6];
  const float* Wd1 = (const float*)d_in[7];
  const float* bd1 = (const float*)d_in[8];
  const float* Wd2 = (const float*)d_in[9];
  const float* bd2 = (const float*)d_in[10];
  const float* Wd3 = (const float*)d_in[11];
  const float* bd3 = (const float*)d_in[12];
  const float* Wemb2 = (const float*)d_in[13];
  const float* bemb2 = (const float*)d_in[14];
  const float* Wemb3 = (const float*)d_in[15];
  const float* bemb3 = (const float*)d_in[16];
  const float* Wt0 = (const float*)d_in[17];
  const float* Wt1 = (const float*)d_in[18];
  const float* Wt2 = (const float*)d_in[19];
  const float* Ws1 = (const float*)d_in[20];
  const float* bs1 = (const float*)d_in[21];
  const float* Ws2 = (const float*)d_in[22];
  const float* bs2 = (const float*)d_in[23];
  const float* ln_g = (const float*)d_in[24];
  const float* ln_b = (const float*)d_in[25];

  const int N = in_sizes[0];        // node_type
  const int E = in_sizes[2];        // bond_dist

  // Workspace: segFI [N*32] | segA [N*32*3] | segS [N*32*5]
  float* ws = (float*)d_ws;
  float* segFI = ws;
  float* segA = ws + (size_t)N * UDIM;
  float* segS = ws + (size_t)N * UDIM * 4;
  const int zn = N * UDIM * 9;
  te_zero_ws<<<(zn + 255) / 256, 256, 0, stream>>>(ws, zn);

  float* X_out = (float*)d_out;                        // [N,32,3,3]
  float* edge_feat = X_out + (size_t)N * UDIM * 9;     // [E,32]

  const long tilesE = ((long)E + 15) / 16;
  const long blocksE = (tilesE + 3) / 4;  // 4 waves/block, 128 threads
  te_edge_kernel<<<(int)blocksE, 128, 0, stream>>>(
      edge_attr, bond_dist, bond_vec, src, dst, node_type, emb_table,
      Wd1, bd1, Wd2, bd2, Wd3, bd3, Wemb2, bemb2, Wemb3, bemb3,
      segFI, segA, segS, edge_feat, E);

  const long tilesN = ((long)N + 15) / 16;
  const long blocksN = (tilesN + 1) / 2;  // 2 waves/block, 64 threads
  te_node_kernel<<<(int)blocksN, 64, 0, stream>>>(
      segFI, segA, segS, Wt0, Wt1, Wt2, Ws1, bs1, Ws2, bs2, ln_g, ln_b,
      X_out, N);
}